// SRNN_34737695490737
// MI455X (gfx1250) — compile-verified
//
#include <hip/hip_runtime.h>
#include <math.h>

// ---------------------------------------------------------------------------
// SRNN: x += DT*(-x + J@rates + inp); rates = 0.5*(1+tanh(x)); out = w@rates/N
// MI455X: J (16MB) is L2-resident (192MB L2); per-step fp32 GEMM via
// V_WMMA_F32_16X16X4_F32.  Pipelined double-buffered LDS GEMM; each wave owns
// a 16x32 C slab (2 accumulators sharing the A fragment -> 2 independent WMMA
// chains).  Rates tile stored K-pair-major in LDS so every A and B fragment
// is one aligned ds_load_b64 (no register repacking in the inner loop).
// ---------------------------------------------------------------------------

typedef __attribute__((ext_vector_type(2))) float v2f;
typedef __attribute__((ext_vector_type(8))) float v8f;

#define SRNN_DT 0.1f
#define SRNN_ON_TIME 10

// Block macro tile: 64(M) x 64(P); 8 waves arranged 4(M) x 2(P-groups),
// each wave computes 16(M) x 32(P).  K blocked by 32, double-buffered LDS.
#define KB   32
#define LDJ  36   // Jt row stride (floats): rows 16B-aligned, conflict-free A reads
#define LDR2 160  // Rt k-pair row stride (words): 8B-aligned pairs; +LDR2 lands
                  // on banks +32 -> half-waves use disjoint bank sets

__global__ __launch_bounds__(256) void srnn_step_kernel(
    const float* __restrict__ J,        // [N, N] row-major
    const float* __restrict__ patterns, // [N, P] row-major
    float* __restrict__ x,              // [N, P] state (in/out)
    const float* __restrict__ r_in,     // [N, P] rates(t-1)
    float* __restrict__ r_out,          // [N, P] rates(t)
    int N, int P, int t)
{
    __shared__ float Jt[2][64 * LDJ];          // J tile, row-major
    __shared__ float Rt[2][(KB / 2) * LDR2];   // rates tile, k-pair-major:
                                               // (k,p) -> [k>>1][p*2 + (k&1)]

    const int tid  = threadIdx.x;
    const int lane = tid & 31;
    const int wave = tid >> 5;       // 0..7
    const int wm   = wave & 3;       // wave tile row in M (0..3)
    const int wp   = wave >> 2;      // wave P-group (0..1), 32 cols each
    const int m0   = blockIdx.x * 64;
    const int p0   = blockIdx.y * 64;
    const int half = lane >> 4;      // 0: lanes 0-15, 1: lanes 16-31
    const int l16  = lane & 15;

    // Cooperative-load coordinates.
    // J tile: 64x32 floats = 512 float4, 2 per thread.
    const int jrow0 = tid >> 3,         jc0 = (tid & 7) * 4;
    const int jrow1 = (tid + 256) >> 3, jc1 = jc0;
    // R tile: 32(K) x 64(P); each thread owns one k-pair x 4 columns.
    const int rkp = tid >> 4;           // k-pair 0..15
    const int rp4 = (tid & 15) * 4;     // column offset 0..60

    v8f acc0 = {0.f, 0.f, 0.f, 0.f, 0.f, 0.f, 0.f, 0.f};
    v8f acc1 = {0.f, 0.f, 0.f, 0.f, 0.f, 0.f, 0.f, 0.f};

    float4 jr0, jr1, ra, rb;

    // Prologue: fetch K-block 0 and stage into LDS buffer 0.
    jr0 = *(const float4*)(J + (size_t)(m0 + jrow0) * N + jc0);
    jr1 = *(const float4*)(J + (size_t)(m0 + jrow1) * N + jc1);
    ra  = *(const float4*)(r_in + (size_t)(2 * rkp)     * P + p0 + rp4);
    rb  = *(const float4*)(r_in + (size_t)(2 * rkp + 1) * P + p0 + rp4);
    *(float4*)(&Jt[0][jrow0 * LDJ + jc0]) = jr0;
    *(float4*)(&Jt[0][jrow1 * LDJ + jc1]) = jr1;
    {
        float* rdst = &Rt[0][rkp * LDR2 + rp4 * 2];
        *(v2f*)(rdst + 0) = (v2f){ra.x, rb.x};
        *(v2f*)(rdst + 2) = (v2f){ra.y, rb.y};
        *(v2f*)(rdst + 4) = (v2f){ra.z, rb.z};
        *(v2f*)(rdst + 6) = (v2f){ra.w, rb.w};
    }
    __syncthreads();

    const int NKB = N / KB;
    for (int kb = 0; kb < NKB; ++kb) {
        const int cur = kb & 1;
        const int nxt = cur ^ 1;
        const bool more = (kb + 1 < NKB);

        // Issue global loads for K-block kb+1 (latency hidden behind WMMAs).
        if (more) {
            const int k0n = (kb + 1) * KB;
            jr0 = *(const float4*)(J + (size_t)(m0 + jrow0) * N + k0n + jc0);
            jr1 = *(const float4*)(J + (size_t)(m0 + jrow1) * N + k0n + jc1);
            ra  = *(const float4*)(r_in + (size_t)(k0n + 2 * rkp)     * P + p0 + rp4);
            rb  = *(const float4*)(r_in + (size_t)(k0n + 2 * rkp + 1) * P + p0 + rp4);
        }

        // WMMA burst: every fragment is one aligned ds_load_b64.
        const float* Jbase = &Jt[cur][(wm * 16 + l16) * LDJ + half * 2];
        const float* Rbase = &Rt[cur][half * LDR2 + (wp * 32 + l16) * 2];
        #pragma unroll
        for (int kk = 0; kk < KB / 4; ++kk) {
            v2f av = *(const v2f*)(Jbase + kk * 4);
            v2f b0 = *(const v2f*)(Rbase + kk * 2 * LDR2);
            v2f b1 = *(const v2f*)(Rbase + kk * 2 * LDR2 + 32);
            acc0 = __builtin_amdgcn_wmma_f32_16x16x4_f32(
                false, av, false, b0, (short)0, acc0, false, false);
            acc1 = __builtin_amdgcn_wmma_f32_16x16x4_f32(
                false, av, false, b1, (short)0, acc1, false, false);
        }

        // Stage K-block kb+1 into the other LDS buffer.
        if (more) {
            *(float4*)(&Jt[nxt][jrow0 * LDJ + jc0]) = jr0;
            *(float4*)(&Jt[nxt][jrow1 * LDJ + jc1]) = jr1;
            float* rdst = &Rt[nxt][rkp * LDR2 + rp4 * 2];
            *(v2f*)(rdst + 0) = (v2f){ra.x, rb.x};
            *(v2f*)(rdst + 2) = (v2f){ra.y, rb.y};
            *(v2f*)(rdst + 4) = (v2f){ra.z, rb.z};
            *(v2f*)(rdst + 6) = (v2f){ra.w, rb.w};
        }
        __syncthreads();  // nxt visible to all; cur reads done before reuse
    }

    // Epilogue: Euler update + shifted/scaled tanh straight from accumulators.
    // C/D layout: VGPR r -> M = r (lanes 0-15) or r+8 (lanes 16-31), N = l16.
    #pragma unroll
    for (int c = 0; c < 2; ++c) {
        const v8f acc = c ? acc1 : acc0;
        const int p = p0 + wp * 32 + c * 16 + l16;
        #pragma unroll
        for (int r = 0; r < 8; ++r) {
            int m = m0 + wm * 16 + r + half * 8;
            size_t idx = (size_t)m * P + p;
            float y   = acc[r];
            float xo  = x[idx];
            float inp = (t < SRNN_ON_TIME) ? patterns[idx] : 0.f;
            float xn  = xo + SRNN_DT * (-xo + y + inp);
            x[idx]    = xn;
            r_out[idx] = 0.5f * (1.0f + tanhf(xn));
        }
    }
}

// Deterministic readout: out[p, t] = (1/N) * sum_n w_out[n] * rates[n, p]
__global__ __launch_bounds__(256) void srnn_readout_kernel(
    const float* __restrict__ w_out,
    const float* __restrict__ rates,   // [N, P]
    float* __restrict__ out,           // [P, T]
    int N, int P, int T, int t)
{
    __shared__ float red[256];
    const int p   = blockIdx.x;
    const int tid = threadIdx.x;
    float s = 0.f;
    for (int n = tid; n < N; n += 256)
        s += w_out[n] * rates[(size_t)n * P + p];
    red[tid] = s;
    __syncthreads();
    #pragma unroll
    for (int off = 128; off > 0; off >>= 1) {
        if (tid < off) red[tid] += red[tid + off];
        __syncthreads();
    }
    if (tid == 0)
        out[(size_t)p * T + t] = red[0] / (float)N;
}

__global__ void srnn_init_kernel(float* __restrict__ x, float* __restrict__ r,
                                 size_t n)
{
    size_t i = (size_t)blockIdx.x * blockDim.x + threadIdx.x;
    if (i < n) {
        x[i] = 0.f;     // x0 = 0
        r[i] = 0.5f;    // rates0 = 0.5*(1+tanh(0))
    }
}

extern "C" void kernel_launch(void* const* d_in, const int* in_sizes, int n_in,
                              void* d_out, int out_size, void* d_ws,
                              size_t ws_size, hipStream_t stream)
{
    const float* patterns = (const float*)d_in[0];  // [N, P]
    const float* J        = (const float*)d_in[1];  // [N, N]
    const float* w_out    = (const float*)d_in[2];  // [N]
    // d_in[3] = N_time_steps lives on device; derive T from sizes instead.

    const int N = in_sizes[2];           // 2048
    const int P = in_sizes[0] / N;       // 256
    const int T = out_size / P;          // 200

    const size_t np = (size_t)N * P;
    float* x  = (float*)d_ws;            // [N*P]
    float* r0 = x + np;                  // [N*P]
    float* r1 = r0 + np;                 // [N*P]  (6 MB total workspace)

    srnn_init_kernel<<<(unsigned)((np + 255) / 256), 256, 0, stream>>>(x, r0, np);

    float* rin  = r0;
    float* rout = r1;
    dim3 grid((unsigned)(N / 64), (unsigned)(P / 64));
    for (int t = 0; t < T; ++t) {
        srnn_step_kernel<<<grid, 256, 0, stream>>>(J, patterns, x, rin, rout,
                                                   N, P, t);
        srnn_readout_kernel<<<(unsigned)P, 256, 0, stream>>>(
            w_out, rout, (float*)d_out, N, P, T, t);
        float* tmp = rin; rin = rout; rout = tmp;
    }
}